// BaselineModel_61478161875493
// MI455X (gfx1250) — compile-verified
//
#include <hip/hip_runtime.h>
#include <math.h>

// ---------------------------------------------------------------------------
// Problem constants (match reference)
// ---------------------------------------------------------------------------
#define BB      8
#define NN      512
#define CC      64
#define BN      4096            // BB*NN
#define OUTSTR  8193            // 1 + 2*BN
#define THR     0.99f
#define INV_T   (1.0f / 0.07f)
#define EPSN    1e-8f

typedef float v2f __attribute__((ext_vector_type(2)));
typedef float v8f __attribute__((ext_vector_type(8)));

// ---------------------------------------------------------------------------
// Kernel 1: row-wise L2 normalize Q/Pos/G, positive logits (col 0), lq gathers
// One wave32 per row; each lane owns 2 of the 64 channels.
// ---------------------------------------------------------------------------
__global__ __launch_bounds__(256) void norm_kernel(
    const float* __restrict__ Q, const float* __restrict__ P,
    const float* __restrict__ G,
    const int* __restrict__ pos_ids, const int* __restrict__ neg_ids,
    const float* __restrict__ logq,
    float* __restrict__ Qn, float* __restrict__ Pn, float* __restrict__ Gn,
    float* __restrict__ lqpos, float* __restrict__ lqneg,
    float* __restrict__ out)
{
    const int row  = (blockIdx.x * blockDim.x + threadIdx.x) >> 5;
    const int lane = threadIdx.x & 31;
    if (row >= BN) return;

    const float2 q = ((const float2*)(Q + (size_t)row * CC))[lane];
    const float2 p = ((const float2*)(P + (size_t)row * CC))[lane];
    const float2 g = ((const float2*)(G + (size_t)row * CC))[lane];

    float sq = q.x * q.x + q.y * q.y;
    float sp = p.x * p.x + p.y * p.y;
    float sg = g.x * g.x + g.y * g.y;
    #pragma unroll
    for (int off = 16; off >= 1; off >>= 1) {
        sq += __shfl_xor(sq, off);
        sp += __shfl_xor(sp, off);
        sg += __shfl_xor(sg, off);
    }
    const float rq = 1.0f / fmaxf(sqrtf(sq), EPSN);
    const float rp = 1.0f / fmaxf(sqrtf(sp), EPSN);
    const float rg = 1.0f / fmaxf(sqrtf(sg), EPSN);

    float2 qn = make_float2(q.x * rq, q.y * rq);
    float2 pn = make_float2(p.x * rp, p.y * rp);
    float2 gn = make_float2(g.x * rg, g.y * rg);

    ((float2*)(Qn + (size_t)row * CC))[lane] = qn;
    ((float2*)(Pn + (size_t)row * CC))[lane] = pn;
    ((float2*)(Gn + (size_t)row * CC))[lane] = gn;

    // positive logit = dot(Qn,Pn)/TEMP - logq[pos_id]
    float d = qn.x * pn.x + qn.y * pn.y;
    #pragma unroll
    for (int off = 16; off >= 1; off >>= 1) d += __shfl_xor(d, off);

    if (lane == 0) {
        const float lp = logq[pos_ids[row]];
        const float ln = logq[neg_ids[row]];
        lqpos[row] = lp;
        lqneg[row] = ln;
        out[(size_t)row * OUTSTR] = d * INV_T - lp;
    }
}

// ---------------------------------------------------------------------------
// Kernel 2: per-sequence column maxima of Pn@Gn^T and Pn@Pn^T via f32 WMMA.
// One wave per (matrix, seq b, 16-column tile). 32 row-tiles * 16 K-steps.
// A/B fragment pattern (16x16x4 f32): lane L -> row/col (L&15),
// float2 at k-offset (L>>4)*2 within each K=4 step.
// ---------------------------------------------------------------------------
__global__ __launch_bounds__(256) void mask_kernel(
    const float* __restrict__ Pn, const float* __restrict__ Gn,
    float* __restrict__ maxPG, float* __restrict__ maxPP)
{
    const int wid  = (blockIdx.x * blockDim.x + threadIdx.x) >> 5;
    const int lane = threadIdx.x & 31;

    const int mat = wid >> 11;                 // 0: vs Gn, 1: vs Pn
    const int rem = wid & 2047;
    const int b   = rem >> 8;                  // sequence 0..7
    const int jt  = rem & 255;                 // 16-col tile 0..255

    const float* Y      = mat ? Pn : Gn;
    float*       outMax = mat ? maxPP : maxPG;

    const int khalf = (lane >> 4) * 2;
    const int col   = jt * 16 + (lane & 15);

    // B fragments for the full K=64 (16 steps), reused across 32 row tiles
    v2f bf[16];
    const float* yrow = Y + (size_t)col * CC + khalf;
    #pragma unroll
    for (int k = 0; k < 16; ++k) bf[k] = *(const v2f*)(yrow + 4 * k);

    float m = -3.402823e38f;
    const float* Abase = Pn + ((size_t)b * NN) * CC + khalf;

    for (int rt = 0; rt < 32; ++rt) {
        const float* arow = Abase + (size_t)(rt * 16 + (lane & 15)) * CC;
        v8f acc = {};
        #pragma unroll
        for (int k = 0; k < 16; ++k) {
            v2f af = *(const v2f*)(arow + 4 * k);
            acc = __builtin_amdgcn_wmma_f32_16x16x4_f32(
                false, af, false, bf[k], (short)0, acc, false, false);
        }
        #pragma unroll
        for (int r = 0; r < 8; ++r) m = fmaxf(m, acc[r]);
    }
    // combine the two M-halves (rows 0-7 vs 8-15 of each tile)
    m = fmaxf(m, __shfl_xor(m, 16));
    if (lane < 16) outMax[b * BN + col] = m;
}

// ---------------------------------------------------------------------------
// Kernel 3: masked logits. One wave per 64x64 output block = 4x4 independent
// 16x16 WMMA tiles. Per K-step: 8 fragment loads feed 16 independent WMMAs
// (0.5 loads/wmma, 16 independent accumulation chains for the matrix pipe).
// A 64-row block never straddles a sequence (512 % 64 == 0).
// ---------------------------------------------------------------------------
__global__ __launch_bounds__(256) void logits_kernel(
    const float* __restrict__ Qn, const float* __restrict__ Pn,
    const float* __restrict__ Gn,
    const float* __restrict__ maxPG, const float* __restrict__ maxPP,
    const float* __restrict__ lqpos, const float* __restrict__ lqneg,
    float* __restrict__ out)
{
    const int wid  = (blockIdx.x * blockDim.x + threadIdx.x) >> 5;
    const int lane = threadIdx.x & 31;

    const int mat = wid >> 12;                 // 0: global negs, 1: in-batch
    const int rem = wid & 4095;
    const int it  = rem >> 6;                  // 64-row block 0..63
    const int jt  = rem & 63;                  // 64-col block 0..63

    const float* Y = mat ? Pn : Gn;

    const int khalf = (lane >> 4) * 2;
    const int l15   = lane & 15;

    const float* arow0 = Qn + (size_t)(it * 64 + l15) * CC + khalf;
    const float* brow0 = Y  + (size_t)(jt * 64 + l15) * CC + khalf;

    v8f acc[4][4] = {};

    #pragma unroll
    for (int k = 0; k < 16; ++k) {
        v2f af[4], bf[4];
        #pragma unroll
        for (int s = 0; s < 4; ++s) {
            af[s] = *(const v2f*)(arow0 + (size_t)s * 16 * CC + 4 * k);
            bf[s] = *(const v2f*)(brow0 + (size_t)s * 16 * CC + 4 * k);
        }
        #pragma unroll
        for (int i = 0; i < 4; ++i) {
            #pragma unroll
            for (int j = 0; j < 4; ++j) {
                acc[i][j] = __builtin_amdgcn_wmma_f32_16x16x4_f32(
                    false, af[i], false, bf[j], (short)0, acc[i][j],
                    false, false);
            }
        }
    }

    // Whole 64-row block lies in one sequence: iseq = (it*64)/512
    const int    iseq   = it >> 3;
    const float* maxArr = mat ? maxPP : maxPG;
    const float* lqArr  = mat ? lqpos : lqneg;
    const float  NEGI   = -__builtin_inff();

    #pragma unroll
    for (int j = 0; j < 4; ++j) {
        const int   col  = jt * 64 + j * 16 + l15;
        const float mv   = maxArr[iseq * BN + col];
        const float lq   = lqArr[col];
        const int   jseq = col >> 9;
        const bool  ok   = (mv <= THR) && (mat == 0 || iseq != jseq);
        const size_t cidx = (size_t)(1 + mat * BN + col);
        #pragma unroll
        for (int i = 0; i < 4; ++i) {
            const int rowBase = it * 64 + i * 16 + (lane >> 4) * 8;
            #pragma unroll
            for (int r = 0; r < 8; ++r) {
                const float v = ok ? (acc[i][j][r] * INV_T - lq) : NEGI;
                out[(size_t)(rowBase + r) * OUTSTR + cidx] = v;
            }
        }
    }
}

// ---------------------------------------------------------------------------
// Launcher
// ---------------------------------------------------------------------------
extern "C" void kernel_launch(void* const* d_in, const int* in_sizes, int n_in,
                              void* d_out, int out_size, void* d_ws, size_t ws_size,
                              hipStream_t stream)
{
    const float* Q       = (const float*)d_in[0];
    const float* Pos     = (const float*)d_in[1];
    const float* G       = (const float*)d_in[2];
    const int*   pos_ids = (const int*)d_in[3];
    const int*   neg_ids = (const int*)d_in[4];
    const float* logq    = (const float*)d_in[5];
    float*       out     = (float*)d_out;

    float* ws    = (float*)d_ws;
    float* Qn    = ws;                 // 4096*64
    float* Pn    = ws + 262144;        // 4096*64
    float* Gn    = ws + 524288;        // 4096*64
    float* maxPG = ws + 786432;        // 8*4096
    float* maxPP = ws + 819200;        // 8*4096
    float* lqpos = ws + 851968;        // 4096
    float* lqneg = ws + 856064;        // 4096

    // 4096 waves (one per row): 512 blocks * 8 waves
    norm_kernel<<<512, 256, 0, stream>>>(Q, Pos, G, pos_ids, neg_ids, logq,
                                         Qn, Pn, Gn, lqpos, lqneg, out);

    // 2 matrices * 8 seqs * 256 col-tiles = 4096 waves
    mask_kernel<<<512, 256, 0, stream>>>(Pn, Gn, maxPG, maxPP);

    // 2 matrices * 64 * 64 blocks (64x64 each) = 8192 waves
    logits_kernel<<<1024, 256, 0, stream>>>(Qn, Pn, Gn, maxPG, maxPP,
                                            lqpos, lqneg, out);
}